// LNC_62904091018196
// MI455X (gfx1250) — compile-verified
//
#include <hip/hip_runtime.h>
#include <math.h>

#define VN 20000
#define KN 32
#define FN 128
#define THRESH 0.9f
#define GT 16                 // groups per reduce tile
#define OUT_ROW (2 * FN)      // 256 floats per output row
#define OUT_MEAN_SZ ((size_t)VN * OUT_ROW)

typedef __attribute__((ext_vector_type(2))) float v2f;
typedef __attribute__((ext_vector_type(8))) float v8f;

// ---------------------------------------------------------------------------
// Phase 0: sigmoid(score) -> s_sig ; sel = -1 everywhere ; npg = 0
// ---------------------------------------------------------------------------
__global__ __launch_bounds__(256) void k_init(const float* __restrict__ score,
                                              float* __restrict__ s_sig,
                                              int* __restrict__ sel,
                                              int* __restrict__ npg) {
  int i = blockIdx.x * 256 + threadIdx.x;
  if (i < VN) {
    s_sig[i] = 1.0f / (1.0f + expf(-score[i]));
    npg[i] = 0;
  }
  int stride = gridDim.x * 256;
  for (int j = i; j < VN * KN; j += stride) sel[j] = -1;
}

// ---------------------------------------------------------------------------
// Phase 1: stable descending rank-sort: order[rank(i)] = i
// rank(i) = #{ j : s[j] > s[i]  ||  (s[j] == s[i] && j < i) }
// Scores tiled through LDS; 400M compares total across the device.
// ---------------------------------------------------------------------------
#define RTILE 2048
__global__ __launch_bounds__(256) void k_rank(const float* __restrict__ s,
                                              int* __restrict__ order) {
  __shared__ float sh[RTILE];
  int i = blockIdx.x * 256 + threadIdx.x;
  float si = (i < VN) ? s[i] : -1.0f;
  int rank = 0;
  for (int base = 0; base < VN; base += RTILE) {
    int lim = min(RTILE, VN - base);
    __syncthreads();
    for (int j = threadIdx.x; j < lim; j += 256) sh[j] = s[base + j];
    __syncthreads();
    if (i < VN) {
      for (int j = 0; j < lim; ++j) {
        float sj = sh[j];
        rank += ((sj > si) || (sj == si && (base + j) < i)) ? 1 : 0;
      }
    }
  }
  if (i < VN) order[rank] = i;
}

// ---------------------------------------------------------------------------
// Phase 2: greedy exclusive clustering, single wave32 (lane = neighbour k).
// assigned[] lives in LDS (20 KB). Sequential over V in score order.
// ---------------------------------------------------------------------------
__global__ __launch_bounds__(32) void k_group(const int* __restrict__ order,
                                              const float* __restrict__ s_sig,
                                              const int* __restrict__ nidxs,
                                              int* __restrict__ sel,
                                              int* __restrict__ gidx,
                                              int* __restrict__ npg,
                                              int* __restrict__ countp) {
  __shared__ unsigned char assigned[VN];
  int lane = threadIdx.x;
  for (int i = lane; i < VN; i += 32) assigned[i] = 0;
  __syncthreads();
  int count = 0;
  for (int it = 0; it < VN; ++it) {
    int v = order[it];
    bool is_new = (assigned[v] == 0);
    bool above = s_sig[v] >= THRESH;
    int nb = nidxs[v * KN + lane];
    bool valid = nb >= 0;
    int nbs = valid ? nb : 0;
    bool take = is_new && valid && (assigned[nbs] == 0) && (above || nb == v);
    __syncthreads();                 // reads-before-writes within the wave
    if (take) { assigned[nb] = 1; gidx[nb] = count; }
    if (is_new) sel[count * KN + lane] = take ? nb : -1;
    unsigned long long bal = __ballot(take);
    if (is_new && lane == 0) npg[count] = (int)__popcll(bal);
    count += is_new ? 1 : 0;
    __syncthreads();
  }
  if (lane == 0) *countp = count;
}

// ---------------------------------------------------------------------------
// Phase 3: gather + segmented reduce via WMMA.
// Block = 16 groups; wave w = 16-feature slab. One f32 WMMA chain of 128
// steps computes D[16 feat x 16 grp] = feat^T x membership(0/1); max rides
// along on VALU. Each gathered element is loaded exactly once (~320 MB total).
// A layout (16x4 f32): lanes 0-15 -> M=lane, K=0,1 ; lanes 16-31 -> K=2,3.
// B layout (4x16 f32): lanes 0-15 -> N=lane, K=0,1 ; lanes 16-31 -> K=2,3.
// C/D layout: vgpr r -> M=r (lanes 0-15) / M=r+8 (lanes 16-31), N=lane&15.
// ---------------------------------------------------------------------------
__global__ __launch_bounds__(256) void k_reduce(const float* __restrict__ feat,
                                                const int* __restrict__ sel,
                                                const int* __restrict__ npg,
                                                float* __restrict__ out) {
  __shared__ int ssel[GT * KN];   // 2 KB: the tile's member indices
  int g0 = blockIdx.x * GT;
  int t = threadIdx.x;
  for (int i = t; i < GT * KN; i += 256) ssel[i] = sel[g0 * KN + i];
  __syncthreads();

  int wave = t >> 5;              // 0..7 -> feature slab
  int lane = t & 31;
  int n = lane & 15;              // B column / A row (feature) index
  int hi = lane >> 4;             // 0: K=0,1  1: K=2,3
  int fb = wave * 16;

  v8f acc = {0.f, 0.f, 0.f, 0.f, 0.f, 0.f, 0.f, 0.f};

  for (int g = 0; g < GT; ++g) {
    float m0 = -1000.f, m1 = -1000.f;
#pragma unroll
    for (int ts = 0; ts < KN / 4; ++ts) {
      int j0 = ts * 4 + hi * 2;              // this lane's two K-slots
      int k0 = ssel[g * KN + j0];
      int k1 = ssel[g * KN + j0 + 1];
      bool v0 = k0 >= 0, v1 = k1 >= 0;
      float a0 = feat[(size_t)(v0 ? k0 : 0) * FN + fb + n];
      float a1 = feat[(size_t)(v1 ? k1 : 0) * FN + fb + n];
      v2f a; a.x = a0; a.y = a1;
      v2f b;
      b.x = (v0 && n == g) ? 1.0f : 0.0f;    // membership mask (default 0.0)
      b.y = (v1 && n == g) ? 1.0f : 0.0f;
      acc = __builtin_amdgcn_wmma_f32_16x16x4_f32(
          false, a, false, b, (short)0, acc, false, false);
      m0 = fmaxf(m0, v0 ? a0 : -1000.f);     // max path, default -1000
      m1 = fmaxf(m1, v1 ? a1 : -1000.f);
    }
    float mg = fmaxf(m0, m1);
    mg = fmaxf(mg, __shfl_xor(mg, 16, 32));  // combine K-halves (lane ^ 16)
    if (lane < 16)
      out[(size_t)(g0 + g) * OUT_ROW + FN + fb + lane] = mg;
  }

  // mean = sum / (npg + 1e-6); lane holds group g0+n, features fb+hi*8+r
  float np = (float)npg[g0 + n];
  size_t row = (size_t)(g0 + n) * OUT_ROW;
  int fbase = fb + hi * 8;
#pragma unroll
  for (int r = 0; r < 8; ++r)
    out[row + fbase + r] = acc[r] / (np + 1e-6f);
}

// ---------------------------------------------------------------------------
// Phase 4: rs = [0, count], backgather = gidx (as numeric floats)
// ---------------------------------------------------------------------------
__global__ __launch_bounds__(256) void k_tail(const int* __restrict__ gidx,
                                              const int* __restrict__ countp,
                                              float* __restrict__ out) {
  int i = blockIdx.x * 256 + threadIdx.x;
  if (i == 0) {
    out[OUT_MEAN_SZ] = 0.0f;
    out[OUT_MEAN_SZ + 1] = (float)(*countp);
  }
  if (i < VN) out[OUT_MEAN_SZ + 2 + i] = (float)gidx[i];
}

// ---------------------------------------------------------------------------
extern "C" void kernel_launch(void* const* d_in, const int* in_sizes, int n_in,
                              void* d_out, int out_size, void* d_ws, size_t ws_size,
                              hipStream_t stream) {
  const float* features = (const float*)d_in[0];   // [V,128]
  const float* score    = (const float*)d_in[1];   // [V,1]
  // d_in[2] coords (unused), d_in[4] row_splits (unused), d_in[5] tidxs (unused)
  const int* nidxs      = (const int*)d_in[3];     // [V,32]
  float* out = (float*)d_out;

  // workspace layout (all 4-byte slots)
  float* s_sig = (float*)d_ws;                 // VN floats
  int* order   = (int*)d_ws + VN;              // VN ints
  int* gidx    = (int*)d_ws + 2 * VN;          // VN ints
  int* npg     = (int*)d_ws + 3 * VN;          // VN ints
  int* countp  = (int*)d_ws + 4 * VN;          // 1 int
  int* sel     = (int*)d_ws + 4 * VN + 4;      // VN*KN ints

  dim3 blk(256);
  int gV = (VN + 255) / 256;

  k_init  <<<gV, blk, 0, stream>>>(score, s_sig, sel, npg);
  k_rank  <<<gV, blk, 0, stream>>>(s_sig, order);
  k_group <<<1, dim3(32), 0, stream>>>(order, s_sig, nidxs, sel, gidx, npg, countp);
  k_reduce<<<VN / GT, blk, 0, stream>>>(features, sel, npg, out);
  k_tail  <<<gV, blk, 0, stream>>>(gidx, countp, out);
}